// BottleneckBlock_75282186764686
// MI455X (gfx1250) — compile-verified
//
#include <hip/hip_runtime.h>

#define NN 49152            // nodes
#define NE 393216           // edges
#define NB 2                // batch
#define CI 128
#define CM 32
#define CO 128
#define BN_EPS 1e-5f

typedef __attribute__((ext_vector_type(2))) float v2f;
typedef __attribute__((ext_vector_type(8))) float v8f;

// ---------------- edge normalization ----------------
__global__ void deg_kernel(const int* __restrict__ ei, const float* __restrict__ w,
                           float* __restrict__ deg) {
    int e = blockIdx.x * blockDim.x + threadIdx.x;
    if (e < NE) atomicAdd(&deg[ei[e]], w[e]);
}

__global__ void nw_kernel(const int* __restrict__ ei, const float* __restrict__ w,
                          const float* __restrict__ deg, float* __restrict__ nw) {
    int e = blockIdx.x * blockDim.x + threadIdx.x;
    if (e >= NE) return;
    int r = ei[e], c = ei[NE + e];
    float dr = deg[r], dc = deg[c];
    float ir = dr > 0.0f ? rsqrtf(dr) : 0.0f;
    float ic = dc > 0.0f ? rsqrtf(dc) : 0.0f;
    nw[e] = -ir * w[e] * ic;
}

// ---------------- buf = -src (seed for T2 = 2*L*T1 - T0) ----------------
__global__ void negcopy_kernel(float* __restrict__ dst, const float* __restrict__ src, size_t n) {
    size_t i = (size_t)blockIdx.x * blockDim.x + threadIdx.x;
    if (i < n) dst[i] = -src[i];
}

// ---------------- sparse propagation: out[b,dst,:] += scale*nw[e]*x[b,src,:] ----------------
// One wave per (edge,batch): coalesced row gather, f32 atomic scatter (L2-resident).
template<int C>
__global__ void prop_kernel(float* __restrict__ out, const float* __restrict__ x,
                            const int* __restrict__ ei, const float* __restrict__ nw,
                            float scale) {
    int gw   = (blockIdx.x * blockDim.x + threadIdx.x) >> 5;
    int lane = threadIdx.x & 31;
    if (gw >= NE * NB) return;
    int e = gw % NE;
    int b = gw / NE;
    int s = ei[e], d = ei[NE + e];
    float wv = scale * nw[e];
    const float* xs = x   + ((size_t)b * NN + s) * C;
    float*       od = out + ((size_t)b * NN + d) * C;
#pragma unroll
    for (int c = lane; c < C; c += 32)
        atomicAdd(&od[c], wv * xs[c]);
}

// ---------------- Chebyshev GEMM: Z = T0*W[0] + T1*W[1] + T2*W[2] + bias ----------------
// One wave owns a 16-node x 16-cout tile. WMMA F32 16x16x4 (A: 16x4, B: 4x16).
// A layout (32-bit 16x4): lane = M (l&15), VGPR v holds K = 2*(l>>4)+v  -> float2 load.
// B layout mirrored: lane = N (l&15), VGPR v holds K = 2*(l>>4)+v.
// C/D layout: lane l holds col N = l&15, VGPR v holds row M = v + 8*(l>>4).
template<int CIN, int COUT>
__global__ __launch_bounds__(256) void cheb_gemm_kernel(
        const float* __restrict__ T0, const float* __restrict__ T1, const float* __restrict__ T2,
        const float* __restrict__ W,  const float* __restrict__ bias,
        float* __restrict__ Z, float* __restrict__ stats) {
    const int tilesN = COUT / 16;
    int gw   = (blockIdx.x * 256 + threadIdx.x) >> 5;
    int lane = threadIdx.x & 31;
    int nodeTile = gw / tilesN;
    int ct       = gw % tilesN;
    size_t nodeBase = (size_t)nodeTile * 16;
    int coutBase = ct * 16;
    int mn = lane & 15;     // A row / B col
    int kh = lane >> 4;     // K half

    v8f acc = {};
    const float* Ts[3] = {T0, T1, T2};
#pragma unroll
    for (int t = 0; t < 3; ++t) {
        const float* arow = Ts[t] + (nodeBase + mn) * CIN + 2 * kh;
        const float* bcol = W + (size_t)t * CIN * COUT + (size_t)(2 * kh) * COUT + coutBase + mn;
#pragma unroll 2
        for (int k0 = 0; k0 < CIN; k0 += 4) {
            v2f a;
            a.x = arow[k0];
            a.y = arow[k0 + 1];
            v2f b;
            b.x = bcol[(size_t)k0 * COUT];
            b.y = bcol[(size_t)(k0 + 1) * COUT];
            acc = __builtin_amdgcn_wmma_f32_16x16x4_f32(
                /*neg_a=*/false, a, /*neg_b=*/false, b,
                /*c_mod=*/(short)0, acc, /*reuse_a=*/false, /*reuse_b=*/false);
        }
    }

    // epilogue: bias, store Z, BN partial sums (per channel)
    float bv = bias[coutBase + mn];
    float s = 0.0f, q = 0.0f;
#pragma unroll
    for (int v = 0; v < 8; ++v) {
        float z = acc[v] + bv;
        s += z;
        q += z * z;
        Z[(nodeBase + v + 8 * kh) * COUT + coutBase + mn] = z;
    }
    atomicAdd(&stats[coutBase + mn], s);
    atomicAdd(&stats[128 + coutBase + mn], q);
}

// ---------------- BN (training stats) + ReLU ----------------
template<int C>
__global__ void bn_relu_kernel(const float* __restrict__ Z, float* __restrict__ Y,
                               const float* __restrict__ g, const float* __restrict__ be,
                               const float* __restrict__ stats) {
    size_t i = (size_t)blockIdx.x * blockDim.x + threadIdx.x;
    const size_t total = (size_t)NB * NN * C;
    if (i >= total) return;
    int c = (int)(i % C);
    const float cnt = (float)((size_t)NB * NN);
    float mean = stats[c] / cnt;
    float var  = stats[128 + c] / cnt - mean * mean;
    float yv = (Z[i] - mean) * rsqrtf(var + BN_EPS) * g[c] + be[c];
    Y[i] = fmaxf(yv, 0.0f);
}

// final layer: out = relu(x + relu(bn(z)))
template<int C>
__global__ void bn_relu_res_kernel(const float* __restrict__ Z, const float* __restrict__ x,
                                   float* __restrict__ out,
                                   const float* __restrict__ g, const float* __restrict__ be,
                                   const float* __restrict__ stats) {
    size_t i = (size_t)blockIdx.x * blockDim.x + threadIdx.x;
    const size_t total = (size_t)NB * NN * C;
    if (i >= total) return;
    int c = (int)(i % C);
    const float cnt = (float)((size_t)NB * NN);
    float mean = stats[c] / cnt;
    float var  = stats[128 + c] / cnt - mean * mean;
    float yv = (Z[i] - mean) * rsqrtf(var + BN_EPS) * g[c] + be[c];
    yv = fmaxf(yv, 0.0f);
    out[i] = fmaxf(x[i] + yv, 0.0f);
}

extern "C" void kernel_launch(void* const* d_in, const int* in_sizes, int n_in,
                              void* d_out, int out_size, void* d_ws, size_t ws_size,
                              hipStream_t stream) {
    const float* x   = (const float*)d_in[0];
    const int*   ei  = (const int*)  d_in[1];
    const float* ew  = (const float*)d_in[2];
    const float* W1  = (const float*)d_in[3];
    const float* b1  = (const float*)d_in[4];
    const float* g1  = (const float*)d_in[5];
    const float* be1 = (const float*)d_in[6];
    const float* W2  = (const float*)d_in[7];
    const float* b2  = (const float*)d_in[8];
    const float* g2  = (const float*)d_in[9];
    const float* be2 = (const float*)d_in[10];
    const float* W3  = (const float*)d_in[11];
    const float* b3  = (const float*)d_in[12];
    const float* g3  = (const float*)d_in[13];
    const float* be3 = (const float*)d_in[14];
    float* out = (float*)d_out;

    // workspace layout
    char* ws = (char*)d_ws;
    size_t off = 0;
    auto alloc = [&](size_t bytes) -> float* {
        float* p = (float*)(ws + off);
        off = (off + bytes + 255) & ~(size_t)255;
        return p;
    };
    const size_t n128 = (size_t)NB * NN * CI;   // 12.58M floats
    const size_t n32  = (size_t)NB * NN * CM;
    float* deg   = alloc((size_t)NN * 4);
    float* nw    = alloc((size_t)NE * 4);
    float* bufA  = alloc(n128 * 4);   // T1
    float* bufB  = alloc(n128 * 4);   // T2
    float* bufC  = alloc(n128 * 4);   // Z
    float* bufD  = alloc(n32 * 4);    // y (layer outputs 32ch)
    float* stats = alloc(256 * 4);

    const int EB = 256;
    const int propBlocks = (NE * NB * 32) / 256;        // one wave per (edge,batch)
    const int gemmBlocks32  = ((NB * NN / 16) * (CM / 16)) / 8;  // 1536
    const int gemmBlocks128 = ((NB * NN / 16) * (CO / 16)) / 8;  // 6144

    // ---- edge normalization ----
    hipMemsetAsync(deg, 0, (size_t)NN * 4, stream);
    deg_kernel<<<(NE + EB - 1) / EB, EB, 0, stream>>>(ei, ew, deg);
    nw_kernel<<<(NE + EB - 1) / EB, EB, 0, stream>>>(ei, ew, deg, nw);

    // ---- layer 1: 128 -> 32 ----
    hipMemsetAsync(bufA, 0, n128 * 4, stream);
    prop_kernel<CI><<<propBlocks, EB, 0, stream>>>(bufA, x, ei, nw, 1.0f);      // T1 = L x
    negcopy_kernel<<<(unsigned)((n128 + EB - 1) / EB), EB, 0, stream>>>(bufB, x, n128);
    prop_kernel<CI><<<propBlocks, EB, 0, stream>>>(bufB, bufA, ei, nw, 2.0f);   // T2 = 2 L T1 - x
    hipMemsetAsync(stats, 0, 256 * 4, stream);
    cheb_gemm_kernel<CI, CM><<<gemmBlocks32, 256, 0, stream>>>(x, bufA, bufB, W1, b1, bufC, stats);
    bn_relu_kernel<CM><<<(unsigned)((n32 + EB - 1) / EB), EB, 0, stream>>>(bufC, bufD, g1, be1, stats);

    // ---- layer 2: 32 -> 32 ----
    hipMemsetAsync(bufA, 0, n32 * 4, stream);
    prop_kernel<CM><<<propBlocks, EB, 0, stream>>>(bufA, bufD, ei, nw, 1.0f);
    negcopy_kernel<<<(unsigned)((n32 + EB - 1) / EB), EB, 0, stream>>>(bufB, bufD, n32);
    prop_kernel<CM><<<propBlocks, EB, 0, stream>>>(bufB, bufA, ei, nw, 2.0f);
    hipMemsetAsync(stats, 0, 256 * 4, stream);
    cheb_gemm_kernel<CM, CM><<<gemmBlocks32, 256, 0, stream>>>(bufD, bufA, bufB, W2, b2, bufC, stats);
    bn_relu_kernel<CM><<<(unsigned)((n32 + EB - 1) / EB), EB, 0, stream>>>(bufC, bufD, g2, be2, stats);
    // NOTE: bufD is input and output of this BN, but the GEMM (reader of bufD) completed first
    // and bn_relu reads only bufC/stats -> write bufD is safe. Actually write into a fresh region:
    // (kept simple: bn writes bufD after all reads of previous bufD are done)

    // ---- layer 3: 32 -> 128 ----
    hipMemsetAsync(bufA, 0, n32 * 4, stream);
    prop_kernel<CM><<<propBlocks, EB, 0, stream>>>(bufA, bufD, ei, nw, 1.0f);
    negcopy_kernel<<<(unsigned)((n32 + EB - 1) / EB), EB, 0, stream>>>(bufB, bufD, n32);
    prop_kernel<CM><<<propBlocks, EB, 0, stream>>>(bufB, bufA, ei, nw, 2.0f);
    hipMemsetAsync(stats, 0, 256 * 4, stream);
    cheb_gemm_kernel<CM, CO><<<gemmBlocks128, 256, 0, stream>>>(bufD, bufA, bufB, W3, b3, bufC, stats);
    bn_relu_res_kernel<CO><<<(unsigned)((n128 + EB - 1) / EB), EB, 0, stream>>>(bufC, x, out, g3, be3, stats);

    (void)in_sizes; (void)n_in; (void)out_size; (void)ws_size;
}